// DualStreamNetwork_35966056137384
// MI455X (gfx1250) — compile-verified
//
#include <hip/hip_runtime.h>
#include <hip/hip_bf16.h>
#include <math.h>

typedef float v2f __attribute__((ext_vector_type(2)));
typedef float v8f __attribute__((ext_vector_type(8)));

#define PI_F 3.14159265358979323846f

// ---------------------------------------------------------------------------
// conv1: 3->32, 3x3, stride 2, SAME (pad bottom/right only), + bias + SiLU
// in:  [32,3,224,224], w [32,3,3,3], b [32]  ->  out [32,32,112,112]
// ---------------------------------------------------------------------------
__global__ __launch_bounds__(256)
void conv1_silu_kernel(const float* __restrict__ x, const float* __restrict__ w1,
                       const float* __restrict__ b1, float* __restrict__ out)
{
    int idx = blockIdx.x * 256 + threadIdx.x;            // 32*32*112*112 = 12845056
    if (idx >= 32 * 32 * 112 * 112) return;
    int ox = idx % 112; int t = idx / 112;
    int oy = t % 112;   t /= 112;
    int oc = t % 32;    int img = t / 32;
    float acc = b1[oc];
    const float* wb = w1 + oc * 27;
    const float* xb = x + (size_t)img * 3 * 224 * 224;
    #pragma unroll
    for (int ic = 0; ic < 3; ++ic)
        #pragma unroll
        for (int ky = 0; ky < 3; ++ky) {
            int iy = oy * 2 + ky;
            if (iy >= 224) continue;
            #pragma unroll
            for (int kx = 0; kx < 3; ++kx) {
                int ix = ox * 2 + kx;
                if (ix >= 224) continue;
                acc += xb[(ic * 224 + iy) * 224 + ix] * wb[ic * 9 + ky * 3 + kx];
            }
        }
    out[idx] = acc / (1.f + expf(-acc));                 // SiLU
}

// ---------------------------------------------------------------------------
// conv2 as implicit GEMM with V_WMMA_F32_16X16X4_F32, fused bias+SiLU+avgpool.
// h1: [32,32,112,112] (SiLU applied), w2: [1280,32,3,3], b2: [1280]
// feat: [32,1280].  grid (80 m-tiles, 32 images), 128 threads = 4 waves.
//
// N-tile = 2x8 output patch: C column n -> (dy=n>>3, dx=n&7).
// K is PERMUTED identically for A and B:  k'' = icb*72 + (ky*3+kx)*8 + ic
// and the per-wave LDS input tile is laid out [ly(5)][lx(18)][ic(8)]
// (strides 144/8/1). With the 18-step K loop fully unrolled, every WMMA's
// A and B operands are single ds_load_b64 with compile-time immediate
// offsets: no offset table, no address VALU, no divergence.
// ---------------------------------------------------------------------------
__global__ __launch_bounds__(128)
void conv2_wmma_pool_kernel(const float* __restrict__ h1, const float* __restrict__ w2,
                            const float* __restrict__ b2, float* __restrict__ feat)
{
    __shared__ float sW[16 * 288];          // 18 KB, K-permuted weight tile [m][k'']
    __shared__ float sT[4][720];            // per-wave tile [ly][lx][ic] = 720 floats
    __shared__ float sAcc[16];

    const int mtile = blockIdx.x;           // 0..79
    const int img   = blockIdx.y;           // 0..31
    const int tid   = threadIdx.x;
    const int lane  = tid & 31;
    const int wave  = tid >> 5;
    const int half  = lane >> 4;            // 0: lanes 0-15, 1: lanes 16-31
    const int nl    = lane & 15;

    // Stage weights with the K permutation baked in.
    for (int i = tid; i < 16 * 288; i += 128) {
        int m  = i / 288, kk = i - m * 288;
        int icb = kk / 72, r = kk - icb * 72;
        int s = r >> 3, ic = r & 7;                     // s = ky*3+kx
        sW[i] = w2[(mtile * 16 + m) * 288 + (icb * 8 + ic) * 9 + s];
    }
    if (tid < 16) sAcc[tid] = 0.f;
    __syncthreads();

    // C column n = nl  ->  (dy, dx) inside the 2x8 output patch
    const int dy = nl >> 3, dx = nl & 7;
    const int laneOff = dy * 288 + dx * 16;             // (dy*2)*144 + (dx*2)*8

    // fixed staging slots (ly,lx) for this lane: j, j+32, j+64 within 5x18
    const int j0 = lane, j1 = lane + 32, j2 = lane + 64;
    const int ly0 = j0 / 18, lx0 = j0 - ly0 * 18;
    const int ly1 = j1 / 18, lx1 = j1 - ly1 * 18;
    const int ly2 = j2 / 18, lx2 = j2 - ly2 * 18;
    const bool j2ok = (j2 < 90);
    const int t0 = ly0 * 144 + lx0 * 8;
    const int t1 = ly1 * 144 + lx1 * 8;
    const int t2 = ly2 * 144 + lx2 * 8;

    float bias[8];
    #pragma unroll
    for (int j = 0; j < 8; ++j) bias[j] = b2[mtile * 16 + j + half * 8];

    float accR[8] = {0.f, 0.f, 0.f, 0.f, 0.f, 0.f, 0.f, 0.f};
    const float* base = h1 + (size_t)img * (32 * 112 * 112);
    float* tile = sT[wave];
    const float* tb = tile + laneOff + half * 2;        // B base for this lane

    for (int nt = wave; nt < 196; nt += 4) {            // 28 y-tiles x 7 x-tiles
        int yt = nt / 7, xt = nt - yt * 7;
        int iy0 = yt * 4, ix0 = xt * 16;                // input window origin
        v8f c = {0.f, 0.f, 0.f, 0.f, 0.f, 0.f, 0.f, 0.f};

        for (int icb = 0; icb < 4; ++icb) {             // 8 input channels per stage
            // ---- stage zero-padded 5x17 window, 8 channels (branchless) ----
            #pragma unroll
            for (int ic = 0; ic < 8; ++ic) {
                const float* p = base + (size_t)(icb * 8 + ic) * (112 * 112);
                {
                    int iy = iy0 + ly0, ix = ix0 + lx0;
                    bool ok = (iy < 112) & (ix < 112);
                    float v = p[(iy < 112 ? iy : 111) * 112 + (ix < 112 ? ix : 111)];
                    tile[t0 + ic] = ok ? v : 0.f;
                }
                {
                    int iy = iy0 + ly1, ix = ix0 + lx1;
                    bool ok = (iy < 112) & (ix < 112);
                    float v = p[(iy < 112 ? iy : 111) * 112 + (ix < 112 ? ix : 111)];
                    tile[t1 + ic] = ok ? v : 0.f;
                }
                if (j2ok) {
                    int iy = iy0 + ly2, ix = ix0 + lx2;
                    bool ok = (iy < 112) & (ix < 112);
                    float v = p[(iy < 112 ? iy : 111) * 112 + (ix < 112 ? ix : 111)];
                    tile[t2 + ic] = ok ? v : 0.f;
                }
            }
            // DS ops are in-order within a wave; fence the scheduler + LDS pipe
            asm volatile("s_wait_dscnt 0x0" ::: "memory");

            const float* wrow = sW + nl * 288 + icb * 72 + half * 2;
            #pragma unroll
            for (int kq = 0; kq < 18; ++kq) {           // 18 WMMAs per ic-block
                const int s  = kq >> 1;                 // ky*3+kx (compile-time)
                const int ky = s / 3, kx = s - ky * 3;
                const int bo = ky * 144 + kx * 8 + (kq & 1) * 4;
                v2f a = *(const v2f*)(wrow + kq * 4);   // ds_load_b64, imm offset
                v2f b = *(const v2f*)(tb + bo);         // ds_load_b64, imm offset
                c = __builtin_amdgcn_wmma_f32_16x16x4_f32(false, a, false, b,
                                                          (short)0, c, false, false);
            }
        }
        // Epilogue: C[j] = (M = j + 8*half, N = nl); bias + SiLU, pool over N
        #pragma unroll
        for (int j = 0; j < 8; ++j) {
            float v = c[j] + bias[j];
            accR[j] += v / (1.f + expf(-v));
        }
    }
    #pragma unroll
    for (int j = 0; j < 8; ++j)
        atomicAdd(&sAcc[j + half * 8], accR[j]);        // ds_add_f32
    __syncthreads();
    if (tid < 16)
        feat[img * 1280 + mtile * 16 + tid] = sAcc[tid] * (1.f / 3136.f);
}

// ---------------------------------------------------------------------------
// Frequency path: separable 224-point DFTs (tables in LDS, incremental index)
// ---------------------------------------------------------------------------
__global__ __launch_bounds__(224)
void dft_rows_kernel(const float* __restrict__ x, float* __restrict__ F1)
{
    __shared__ float ct[224], st[224];
    int y  = blockIdx.x % 224;
    int bc = blockIdx.x / 224;                           // img*3+ch, 0..95
    int u  = threadIdx.x;
    float ang = (2.0f * PI_F / 224.0f) * (float)u;
    ct[u] = cosf(ang); st[u] = sinf(ang);
    __syncthreads();
    const float* row = x + ((size_t)bc * 224 + y) * 224;
    float re = 0.f, im = 0.f; int idx = 0;
    for (int xx = 0; xx < 224; ++xx) {
        float v = row[xx];
        re += v * ct[idx]; im -= v * st[idx];            // * exp(-i*2pi*u*x/N)
        idx += u; if (idx >= 224) idx -= 224;
    }
    float* o = F1 + (((size_t)bc * 224 + y) * 224 + u) * 2;
    o[0] = re; o[1] = im;
}

__global__ __launch_bounds__(224)
void dft_cols_mag_kernel(const float* __restrict__ F1, float* __restrict__ mag)
{
    __shared__ float ct[224], st[224];
    int u  = blockIdx.x % 224;
    int bc = blockIdx.x / 224;
    int v  = threadIdx.x;
    float ang = (2.0f * PI_F / 224.0f) * (float)v;
    ct[v] = cosf(ang); st[v] = sinf(ang);
    __syncthreads();
    const float* base = F1 + (size_t)bc * 224 * 224 * 2 + u * 2;
    float re = 0.f, im = 0.f; int idx = 0;
    for (int y = 0; y < 224; ++y) {
        float fr = base[(size_t)y * 448], fi = base[(size_t)y * 448 + 1];
        float cc = ct[idx], ss = st[idx];
        re += fr * cc + fi * ss;                         // (fr+i fi)*(cc - i ss)
        im += fi * cc - fr * ss;
        idx += v; if (idx >= 224) idx -= 224;
    }
    float m = logf(sqrtf(re * re + im * im) + 1.0f);
    int vo = v + 112; if (vo >= 224) vo -= 224;          // fftshift
    int uo = u + 112; if (uo >= 224) uo -= 224;
    mag[((size_t)bc * 224 + vo) * 224 + uo] = m;
}

__global__ __launch_bounds__(256)
void minmax_kernel(const float* __restrict__ mag, float* __restrict__ mm)
{
    __shared__ float smn[256], smx[256];
    int bc = blockIdx.x, tid = threadIdx.x;
    const float* p = mag + (size_t)bc * 50176;
    float mn = 3.0e38f, mx = -3.0e38f;
    for (int i = tid; i < 50176; i += 256) {
        float v = p[i]; mn = fminf(mn, v); mx = fmaxf(mx, v);
    }
    smn[tid] = mn; smx[tid] = mx; __syncthreads();
    for (int s = 128; s > 0; s >>= 1) {
        if (tid < s) { smn[tid] = fminf(smn[tid], smn[tid + s]);
                       smx[tid] = fmaxf(smx[tid], smx[tid + s]); }
        __syncthreads();
    }
    if (tid == 0) { mm[bc * 2] = smn[0]; mm[bc * 2 + 1] = smx[0]; }
}

__global__ __launch_bounds__(256)
void normalize_kernel(float* __restrict__ mag, const float* __restrict__ mm)
{
    int idx = blockIdx.x * 256 + threadIdx.x;
    if (idx >= 96 * 50176) return;
    int bc = idx / 50176;
    float mn = mm[bc * 2], mx = mm[bc * 2 + 1];
    mag[idx] = (mag[idx] - mn) / (mx - mn + 1e-8f);
}

// ---------------------------------------------------------------------------
// Generic small FC: out[i, off+j] = act(in[i,:K] . w[j,:K] + b[j]), i in [0,32)
// ---------------------------------------------------------------------------
__global__ __launch_bounds__(256)
void fc_kernel(const float* __restrict__ in, const float* __restrict__ w,
               const float* __restrict__ b, float* __restrict__ out,
               int K, int N, int outStride, int outOffset, int act)
{
    int idx = blockIdx.x * 256 + threadIdx.x;
    if (idx >= 32 * N) return;
    int i = idx / N, j = idx % N;
    const float* ir = in + (size_t)i * K;
    const float* wr = w + (size_t)j * K;
    float acc = b[j];
    for (int k = 0; k < K; ++k) acc += ir[k] * wr[k];
    if (act == 1) acc = fmaxf(acc, 0.f);
    out[(size_t)i * outStride + outOffset + j] = acc;
}

// ---------------------------------------------------------------------------
// Per-head cross-attention over the batch axis (seq = 32, head dim = 32).
// blockIdx.x = a*8 + h ; a=0: Q from rgb, K/V from frq ; a=1: swapped.
// ---------------------------------------------------------------------------
__global__ __launch_bounds__(256)
void attn_core_kernel(const float* __restrict__ qkv_rgb, const float* __restrict__ qkv_frq,
                      float* __restrict__ o1, float* __restrict__ o2)
{
    __shared__ float sQ[32][32], sK[32][32], sV[32][32], sS[32][33];
    int a = blockIdx.x >> 3;
    int h = blockIdx.x & 7;
    const float* qsrc  = (a == 0) ? qkv_rgb : qkv_frq;
    const float* kvsrc = (a == 0) ? qkv_frq : qkv_rgb;
    float* odst        = (a == 0) ? o1 : o2;
    int tid = threadIdx.x;
    for (int idx = tid; idx < 1024; idx += 256) {
        int q = idx >> 5, d = idx & 31;
        sQ[q][d] = qsrc [q * 768 +       h * 32 + d];
        sK[q][d] = kvsrc[q * 768 + 256 + h * 32 + d];
        sV[q][d] = kvsrc[q * 768 + 512 + h * 32 + d];
    }
    __syncthreads();
    const float scale = 0.17677669529663687f;            // 1/sqrt(32)
    for (int idx = tid; idx < 1024; idx += 256) {
        int q = idx >> 5, k = idx & 31;
        float s = 0.f;
        #pragma unroll
        for (int d = 0; d < 32; ++d) s += sQ[q][d] * sK[k][d];
        sS[q][k] = s * scale;
    }
    __syncthreads();
    if (tid < 32) {
        float mx = -3.0e38f;
        for (int k = 0; k < 32; ++k) mx = fmaxf(mx, sS[tid][k]);
        float e[32], sum = 0.f;
        for (int k = 0; k < 32; ++k) { e[k] = expf(sS[tid][k] - mx); sum += e[k]; }
        float inv = 1.f / sum;
        for (int k = 0; k < 32; ++k) sS[tid][k] = e[k] * inv;
    }
    __syncthreads();
    for (int idx = tid; idx < 1024; idx += 256) {
        int q = idx >> 5, d = idx & 31;
        float o = 0.f;
        #pragma unroll
        for (int k = 0; k < 32; ++k) o += sS[q][k] * sV[k][d];
        odst[q * 256 + h * 32 + d] = o;
    }
}

// ---------------------------------------------------------------------------
extern "C" void kernel_launch(void* const* d_in, const int* in_sizes, int n_in,
                              void* d_out, int out_size, void* d_ws, size_t ws_size,
                              hipStream_t stream)
{
    (void)in_sizes; (void)n_in; (void)out_size; (void)ws_size;
    const float* x          = (const float*)d_in[0];
    const float* rgb_w1     = (const float*)d_in[1];
    const float* rgb_b1     = (const float*)d_in[2];
    const float* rgb_w2     = (const float*)d_in[3];
    const float* rgb_b2     = (const float*)d_in[4];
    const float* frq_w1     = (const float*)d_in[5];
    const float* frq_b1     = (const float*)d_in[6];
    const float* frq_w2     = (const float*)d_in[7];
    const float* frq_b2     = (const float*)d_in[8];
    const float* rgb_proj_w = (const float*)d_in[9];
    const float* rgb_proj_b = (const float*)d_in[10];
    const float* frq_proj_w = (const float*)d_in[11];
    const float* frq_proj_b = (const float*)d_in[12];
    const float* attn_w_in  = (const float*)d_in[13];
    const float* attn_b_in  = (const float*)d_in[14];
    const float* attn_w_out = (const float*)d_in[15];
    const float* attn_b_out = (const float*)d_in[16];
    const float* fus_w1     = (const float*)d_in[17];
    const float* fus_b1     = (const float*)d_in[18];
    const float* fus_w2     = (const float*)d_in[19];
    const float* fus_b2     = (const float*)d_in[20];
    const float* fus_w3     = (const float*)d_in[21];
    const float* fus_b3     = (const float*)d_in[22];

    float* out      = (float*)d_out;                 // [32,2]
    float* feat_rgb = out + 64;                      // [32,1280]
    float* feat_frq = out + 64 + 40960;              // [32,1280]

    // Workspace layout (floats). Region0 is time-shared: h1 (rgb) -> F1 -> h1 (frq)
    float* ws       = (float*)d_ws;
    float* h1       = ws;                            // 12,845,056 (>= F1 9,633,792)
    float* F1       = ws;
    float* freqimg  = ws + 12845056;                 // 4,816,896
    float* mm       = freqimg + 4816896;             // 192
    float* hd       = mm + 192;                      // head scratch
    float* rgb_p    = hd;                            // 8192
    float* frq_p    = rgb_p + 8192;                  // 8192
    float* qkv_rgb  = frq_p + 8192;                  // 24576
    float* qkv_frq  = qkv_rgb + 24576;               // 24576
    float* o1       = qkv_frq + 24576;               // 8192
    float* o2       = o1 + 8192;                     // 8192
    float* hb       = o2 + 8192;                     // 16384 (concat [32,512])
    float* h1f      = hb + 16384;                    // 16384
    float* h2f      = h1f + 16384;                   // 8192

    // ---- RGB stream --------------------------------------------------------
    conv1_silu_kernel<<<50176, 256, 0, stream>>>(x, rgb_w1, rgb_b1, h1);
    conv2_wmma_pool_kernel<<<dim3(80, 32), 128, 0, stream>>>(h1, rgb_w2, rgb_b2, feat_rgb);

    // ---- Frequency stream --------------------------------------------------
    dft_rows_kernel<<<21504, 224, 0, stream>>>(x, F1);
    dft_cols_mag_kernel<<<21504, 224, 0, stream>>>(F1, freqimg);
    minmax_kernel<<<96, 256, 0, stream>>>(freqimg, mm);
    normalize_kernel<<<18816, 256, 0, stream>>>(freqimg, mm);
    conv1_silu_kernel<<<50176, 256, 0, stream>>>(freqimg, frq_w1, frq_b1, h1);
    conv2_wmma_pool_kernel<<<dim3(80, 32), 128, 0, stream>>>(h1, frq_w2, frq_b2, feat_frq);

    // ---- Head --------------------------------------------------------------
    fc_kernel<<<32, 256, 0, stream>>>(feat_rgb, rgb_proj_w, rgb_proj_b, rgb_p, 1280, 256, 256, 0, 0);
    fc_kernel<<<32, 256, 0, stream>>>(feat_frq, frq_proj_w, frq_proj_b, frq_p, 1280, 256, 256, 0, 0);
    fc_kernel<<<96, 256, 0, stream>>>(rgb_p, attn_w_in, attn_b_in, qkv_rgb, 256, 768, 768, 0, 0);
    fc_kernel<<<96, 256, 0, stream>>>(frq_p, attn_w_in, attn_b_in, qkv_frq, 256, 768, 768, 0, 0);
    attn_core_kernel<<<16, 256, 0, stream>>>(qkv_rgb, qkv_frq, o1, o2);
    fc_kernel<<<32, 256, 0, stream>>>(o1, attn_w_out, attn_b_out, hb, 256, 256, 512, 0, 0);
    fc_kernel<<<32, 256, 0, stream>>>(o2, attn_w_out, attn_b_out, hb, 256, 256, 512, 256, 0);
    fc_kernel<<<64, 256, 0, stream>>>(hb,  fus_w1, fus_b1, h1f, 512, 512, 512, 0, 1);
    fc_kernel<<<32, 256, 0, stream>>>(h1f, fus_w2, fus_b2, h2f, 512, 256, 256, 0, 1);
    fc_kernel<<<1,  256, 0, stream>>>(h2f, fus_w3, fus_b3, out, 256, 2, 2, 0, 0);
}